// CIoULoss_15006615733672
// MI455X (gfx1250) — compile-verified
//
#include <hip/hip_runtime.h>

// CIoU loss, fp32, bandwidth-bound streaming + deterministic hierarchical reduce.
// Wave-level reduction via V_WMMA_F32_16X16X4_F32 (gfx1250, wave32).

#define EPS 1e-6f
#define BLOCK 256
#define EPT 4          // elements per thread in the streaming kernel
#define WAVES_PER_BLOCK (BLOCK / 32)

typedef float v2f __attribute__((ext_vector_type(2)));
typedef float v8f __attribute__((ext_vector_type(8)));

__device__ __forceinline__ float fast_rcp(float x) {
    return __builtin_amdgcn_rcpf(x);   // v_rcp_f32, ~1 ulp
}

// One WMMA reduces 32 lane partials to two half-sums:
//   A (16x4): lanes 0-15 VGPR0 -> (M=lane,K=0); lanes 16-31 VGPR0 -> (M=lane-16,K=2)
//   with a = {s, 0}:  A[m][0]=s[m], A[m][2]=s[m+16], rest 0.
//   B = ones(4x16)  =>  D[m][n] = s[m] + s[m+16] for every n.
//   D layout: VGPR r, lanes 0-15 hold M=r, lanes 16-31 hold M=8+r, so the
//   per-lane sum of the 8 D VGPRs gives:
//     lanes 0-15 : sum of s[0..7]  + s[16..23]   (identical across lanes 0-15)
//     lanes 16-31: sum of s[8..15] + s[24..31]   (identical across lanes 16-31)
__device__ __forceinline__ float wave_half_sum_wmma(float s) {
    v2f a; a[0] = s;    a[1] = 0.0f;
    v2f b; b[0] = 1.0f; b[1] = 1.0f;
    v8f c = {};
    v8f d = __builtin_amdgcn_wmma_f32_16x16x4_f32(
        /*neg_a=*/false, a, /*neg_b=*/false, b,
        /*c_mod=*/(short)0, c, /*reuse_a=*/false, /*reuse_b=*/false);
    return ((d[0] + d[1]) + (d[2] + d[3])) + ((d[4] + d[5]) + (d[6] + d[7]));
}

// Block reduce: each wave contributes two half-sums to LDS; thread 0 sums 2*waves.
__device__ __forceinline__ void block_reduce_store(float acc, float* dst, float scale) {
    float r = wave_half_sum_wmma(acc);
    __shared__ float lds[2 * WAVES_PER_BLOCK];
    const int lane = threadIdx.x & 31;
    if ((lane & 15) == 0)
        lds[((threadIdx.x >> 5) << 1) + (lane >> 4)] = r;
    __syncthreads();
    if (threadIdx.x == 0) {
        float s = 0.0f;
#pragma unroll
        for (int w = 0; w < 2 * WAVES_PER_BLOCK; ++w) s += lds[w];
        *dst = s * scale;
    }
}

__device__ __forceinline__ float ciou_loss(float4 P, float4 T) {
    const float c4pi2 = 0.40528473456935108577f;   // 4 / pi^2
    float pw = P.z - P.x, ph = P.w - P.y;
    float tw = T.z - T.x, th = T.w - T.y;
    // intersection / union / IoU
    float ix = fmaxf(fminf(P.z, T.z) - fmaxf(P.x, T.x), 0.0f);
    float iy = fmaxf(fminf(P.w, T.w) - fmaxf(P.y, T.y), 0.0f);
    float inter = ix * iy;
    float uni   = pw * ph + tw * th - inter;
    float iou   = inter * fast_rcp(fmaxf(uni, EPS));
    // squared center distance (centers are 0.5*(lo+hi))
    float dcx = (P.x + P.z) - (T.x + T.z);
    float dcy = (P.y + P.w) - (T.y + T.w);
    float cdist = 0.25f * (dcx * dcx + dcy * dcy);
    // enclosing-box diagonal
    float ex = fmaxf(P.z, T.z) - fminf(P.x, T.x);
    float ey = fmaxf(P.w, T.w) - fminf(P.y, T.y);
    float ddist = ex * ex + ey * ey;
    // aspect term: atan(tw/th) - atan(pw/ph) = atan((tw*ph - pw*th)/(th*ph + tw*pw))
    // valid since tw,th,pw,ph > 0 (data guarantees w,h >= 1e-3 >> EPS)
    float num = tw * ph - pw * th;
    float den = th * ph + tw * pw;   // strictly positive
    float da  = atanf(num * fast_rcp(den));
    float v     = c4pi2 * da * da;
    float alpha = v * fast_rcp((1.0f - iou) + v + EPS);
    float ciou  = iou - cdist * fast_rcp(fmaxf(ddist, EPS)) + alpha * v;
    return 1.0f - ciou;
}

__global__ __launch_bounds__(BLOCK) void ciou_partial_kernel(
        const float4* __restrict__ p, const float4* __restrict__ t,
        float* __restrict__ partial, int n) {
    const int stride = gridDim.x * BLOCK;
    const int tid    = blockIdx.x * BLOCK + threadIdx.x;
    float acc = 0.0f;
#pragma unroll
    for (int k = 0; k < EPT; ++k) {
        int i = tid + k * stride;
        if (i < n) acc += ciou_loss(p[i], t[i]);   // reconverges before WMMA
    }
    block_reduce_store(acc, &partial[blockIdx.x], 1.0f);
}

__global__ __launch_bounds__(BLOCK) void ciou_final_kernel(
        const float* __restrict__ partial, int num_partials,
        float* __restrict__ out, float inv_n) {
    float acc = 0.0f;
    for (int i = threadIdx.x; i < num_partials; i += BLOCK)
        acc += partial[i];                         // reconverges before WMMA
    block_reduce_store(acc, out, inv_n);
}

extern "C" void kernel_launch(void* const* d_in, const int* in_sizes, int n_in,
                              void* d_out, int out_size, void* d_ws, size_t ws_size,
                              hipStream_t stream) {
    const float4* p = (const float4*)d_in[0];
    const float4* t = (const float4*)d_in[1];
    float* out      = (float*)d_out;
    float* partial  = (float*)d_ws;

    const int n = in_sizes[0] / 4;                      // boxes (N = 4,194,304)
    int grid = (n + BLOCK * EPT - 1) / (BLOCK * EPT);   // 4096 for N = 4M
    if (grid < 1) grid = 1;

    ciou_partial_kernel<<<grid, BLOCK, 0, stream>>>(p, t, partial, n);
    ciou_final_kernel<<<1, BLOCK, 0, stream>>>(partial, grid, out, 1.0f / (float)n);
}